// GausSplatingHead_1881195676113
// MI455X (gfx1250) — compile-verified
//
#include <hip/hip_runtime.h>

#define HH   512
#define WW   1408
#define CCH  17
#define HWPIX (HH * WW)

typedef __attribute__((ext_vector_type(2))) float v2f;
typedef __attribute__((ext_vector_type(8))) float v8f;

// ---------------------------------------------------------------------------
// Zero a float buffer (grid-stride).
// ---------------------------------------------------------------------------
__global__ void zero_f32(float* __restrict__ p, int n) {
  int i = blockIdx.x * blockDim.x + threadIdx.x;
  int stride = gridDim.x * blockDim.x;
  for (; i < n; i += stride) p[i] = 0.0f;
}

// ---------------------------------------------------------------------------
// Splat kernel: one thread = one point. Projection of 16-point tiles is done
// with V_WMMA_F32_16X16X4_F32: A = 16x4 homogeneous points, B = 4x16 holding
// [R row0 | R row1 | R row2 | fx*row0+cx*row2 | fy*row1+cy*row2] columns, so
// D columns 2..4 give (pz, u*z, v*z) per point. D is column-per-lane, so the
// three needed columns are transposed through a wave-private LDS tile.
// ---------------------------------------------------------------------------
__global__ __launch_bounds__(256)
void splat_kernel(const float* __restrict__ xyz,
                  const float* __restrict__ feats,
                  const float* __restrict__ density,
                  const float* __restrict__ viewmats,
                  const float* __restrict__ Ks,
                  float* __restrict__ img,
                  int npts, int cid)
{
  __shared__ float xpose[8][2][16][4];   // [wave][group][point-in-group][pz,un,vn]

  const int lane = threadIdx.x & 31;
  const int wav  = threadIdx.x >> 5;
  const int col  = lane & 15;
  const bool hi  = lane >= 16;

  const float* VM = viewmats + cid * 16;   // 4x4 row-major
  const float* KK = Ks + cid * 9;          // 3x3 row-major
  const float fx = KK[0], fy = KK[4], cx = KK[2], cy = KK[5];

  // This wave owns 32 consecutive points.
  const int base = blockIdx.x * 256 + wav * 32;
  const int p    = base + lane;
  const bool pvalid = p < npts;
  const int pc = pvalid ? p : (npts - 1);

  // Issue the prefetch for this point's 17-channel feature line early so it
  // overlaps the WMMA + LDS transpose phase (lowers to global_prefetch_b8).
  __builtin_prefetch(feats + (size_t)pc * CCH, 0, 0);

  // --- Build B matrix operand (uniform per camera). Layout for 4x16 f32 B:
  // VGPR r, lanes 0-15: K=r, N=lane; lanes 16-31: K=2+r, N=lane-16.
  auto bij = [&](int k, int n) -> float {
    if (n > 4) return 0.0f;
    float a0 = VM[0 * 4 + k];   // row0 coeff of input k  (k==3 -> t component)
    float a1 = VM[1 * 4 + k];
    float a2 = VM[2 * 4 + k];
    switch (n) {
      case 0:  return a0;                 // px
      case 1:  return a1;                 // py
      case 2:  return a2;                 // pz
      case 3:  return fx * a0 + cx * a2;  // u*z numerator
      default: return fy * a1 + cy * a2;  // v*z numerator
    }
  };
  const int kbase = hi ? 2 : 0;
  v2f bm;
  bm.x = bij(kbase + 0, col);
  bm.y = bij(kbase + 1, col);

  // --- Two WMMAs cover the wave's 32 points (16 per matrix-M).
  #pragma unroll
  for (int g = 0; g < 2; ++g) {
    int pm = base + g * 16 + col;
    pm = (pm < npts) ? pm : (npts - 1);
    // A layout 16x4 f32: lanes 0-15 hold K=0,1; lanes 16-31 hold K=2,3; M=lane%16.
    v2f a;
    if (!hi) { a.x = xyz[3 * pm + 0]; a.y = xyz[3 * pm + 1]; }
    else     { a.x = xyz[3 * pm + 2]; a.y = 1.0f;            }
    v8f c0 = {};
    v8f d = __builtin_amdgcn_wmma_f32_16x16x4_f32(
        /*neg_a=*/false, a, /*neg_b=*/false, bm,
        /*c_mod=*/(short)0, c0, /*reuse_a=*/false, /*reuse_b=*/false);
    // D layout: VGPR r, lanes 0-15 -> D[M=r][N=lane]; lanes 16-31 -> D[M=8+r][N=lane-16].
    // Stash the three useful columns (N=2,3,4) to LDS for the transpose.
    if (col >= 2 && col < 5) {
      #pragma unroll
      for (int r = 0; r < 8; ++r) {
        int row = hi ? (8 + r) : r;
        xpose[wav][g][row][col - 2] = d[r];
      }
    }
  }
  // Wave-private LDS slice + in-order DS per wave: no barrier needed.

  // --- Gather this point's 17 features + opacity (prefetched above).
  float f[CCH];
  #pragma unroll
  for (int ci = 0; ci < CCH; ++ci) f[ci] = feats[(size_t)pc * CCH + ci];
  const float op = density[pc];

  // --- Read back this lane's projected quantities (transpose).
  const int g2 = lane >> 4;
  const float P2 = xpose[wav][g2][col][0];   // pz
  const float UN = xpose[wav][g2][col][1];   // u*z
  const float VN = xpose[wav][g2][col][2];   // v*z

  const float z  = fmaxf(P2, 0.001f);
  const float iz = 1.0f / z;
  const float u  = UN * iz;                  // == fx*px/z + cx for valid pts
  const float v  = VN * iz;

  // Conic from projected Jacobian (j02 = -(u-cx)/z, j12 = -(v-cy)/z).
  const float j00 = fx * iz, j11 = fy * iz;
  const float j02 = (cx - u) * iz;
  const float j12 = (cy - v) * iz;
  const float s2  = 0.16f;                   // VOXEL_SIZE^2
  const float A  = s2 * (j00 * j00 + j02 * j02) + 0.3f;
  const float Bq = s2 * (j02 * j12);
  const float Cq = s2 * (j11 * j11 + j12 * j12) + 0.3f;
  const float idet = 1.0f / (A * Cq - Bq * Bq);
  const float ca = Cq * idet, cb = -Bq * idet, cc = A * idet;

  const float ru = rintf(u), rv = rintf(v);  // round-half-even like jnp.round
  const bool zok = pvalid && (P2 > 0.1f);

  // --- 3x3 footprint scatter: L2-resident f32 atomics.
  #pragma unroll
  for (int i = 0; i < 3; ++i) {
    #pragma unroll
    for (int j = 0; j < 3; ++j) {
      float pxr = ru + (float)(i - 1);
      float pyr = rv + (float)(j - 1);
      if (zok && pxr >= 0.0f && pxr < (float)WW && pyr >= 0.0f && pyr < (float)HH) {
        float ddx = pxr - u, ddy = pyr - v;
        float gz = __expf(-0.5f * (ca * ddx * ddx + 2.0f * cb * ddx * ddy + cc * ddy * ddy));
        float w = op * gz;
        float* dst = img + (size_t)((int)pyr * WW + (int)pxr) * CCH;
        #pragma unroll
        for (int ci = 0; ci < CCH; ++ci) atomicAdd(dst + ci, w * f[ci]);
      }
    }
  }
}

// ---------------------------------------------------------------------------
// Per-pixel weighted softmax-CE, wave32 shuffle reduction, 2 atomics / wave.
// ---------------------------------------------------------------------------
__global__ __launch_bounds__(256)
void loss_kernel(const float* __restrict__ img,
                 const int* __restrict__ gt,
                 const float* __restrict__ cw,
                 float* __restrict__ accum)  // accum[0]=sum(w*nll), accum[1]=sum(w)
{
  int pix = blockIdx.x * blockDim.x + threadIdx.x;
  float wn = 0.0f, wv = 0.0f;
  if (pix < HWPIX) {
    const float* L = img + (size_t)pix * CCH;
    float m = L[0];
    #pragma unroll
    for (int c = 1; c < CCH; ++c) m = fmaxf(m, L[c]);
    float s = 0.0f;
    #pragma unroll
    for (int c = 0; c < CCH; ++c) s += __expf(L[c] - m);
    float lse = m + __logf(s);
    int g = gt[pix];
    g = (g < 0) ? 0 : ((g >= CCH) ? (CCH - 1) : g);
    float nll = lse - L[g];
    float w = (g != 0) ? cw[g] : 0.0f;
    wn = w * nll;
    wv = w;
  }
  #pragma unroll
  for (int off = 16; off > 0; off >>= 1) {  // wave32 reduction
    wn += __shfl_down(wn, off, 32);
    wv += __shfl_down(wv, off, 32);
  }
  if ((threadIdx.x & 31) == 0) {
    atomicAdd(&accum[0], wn);
    atomicAdd(&accum[1], wv);
  }
}

// ---------------------------------------------------------------------------
// Final scalar: sum_c (wnll_c / max(w_c, 1e-8)) / NC   (B == 1).
// ---------------------------------------------------------------------------
__global__ void finalize_kernel(const float* __restrict__ accum, float* __restrict__ out, int nc) {
  if (threadIdx.x == 0 && blockIdx.x == 0) {
    float s = 0.0f;
    for (int c = 0; c < nc; ++c)
      s += accum[2 * c] / fmaxf(accum[2 * c + 1], 1e-8f);
    out[0] = s / (float)nc;
  }
}

// ---------------------------------------------------------------------------
extern "C" void kernel_launch(void* const* d_in, const int* in_sizes, int n_in,
                              void* d_out, int out_size, void* d_ws, size_t ws_size,
                              hipStream_t stream)
{
  const float* voxel_feats = (const float*)d_in[0];  // (1,200,200,16,17)
  const float* density     = (const float*)d_in[1];  // (1,200,200,16,1)
  const float* viewmats    = (const float*)d_in[2];  // (1,NC,4,4)
  const float* Ks          = (const float*)d_in[3];  // (1,NC,3,3)
  const int*   gt_sem      = (const int*)  d_in[4];  // (1,NC,H,W)
  const float* pc_xyz      = (const float*)d_in[5];  // (npts,3)
  const float* class_w     = (const float*)d_in[6];  // (17,)
  float* out = (float*)d_out;

  const int npts = in_sizes[1];        // #points == density element count
  const int NC   = in_sizes[2] / 16;   // viewmats: NC*4*4

  // Workspace: [0,256) per-camera accumulators; [256, ...) one HWxC logit
  // image (~49 MB) reused across cameras so it stays resident in the 192 MB L2.
  float* accum = (float*)d_ws;
  float* img   = (float*)((char*)d_ws + 256);
  const int HWC = HWPIX * CCH;

  zero_f32<<<1, 64, 0, stream>>>(accum, 2 * NC);

  for (int cid = 0; cid < NC; ++cid) {
    zero_f32<<<2048, 256, 0, stream>>>(img, HWC);
    dim3 grid((npts + 255) / 256);
    splat_kernel<<<grid, 256, 0, stream>>>(pc_xyz, voxel_feats, density,
                                           viewmats, Ks, img, npts, cid);
    loss_kernel<<<(HWPIX + 255) / 256, 256, 0, stream>>>(
        img, gt_sem + (size_t)cid * HWPIX, class_w, accum + 2 * cid);
  }
  finalize_kernel<<<1, 1, 0, stream>>>(accum, out, NC);
}